// ProjectRenderScoreModel_19000935317879
// MI455X (gfx1250) — compile-verified
//
#include <hip/hip_runtime.h>
#include <hip/hip_bf16.h>

#define IMG_H 256
#define NPTS 128
#define NG 6            // bs*3
#define TILE 16
#define HALO 4
#define STILE 24        // TILE + 2*HALO
#define SSTR 26         // LDS row stride (even -> 8B-aligned float2 loads, conflict-free)

typedef float v2f __attribute__((ext_vector_type(2)));
typedef float v8f __attribute__((ext_vector_type(8)));

// ---------------------------------------------------------------------------
// Prep: _normalise_scale_factor for the three (2,3) camera arrays -> d_ws[0..17]
// v = v / mean(v); sf = 0.3/max|v-1|; v = sf<1 ? (v-1)*sf+1 : v
// ---------------------------------------------------------------------------
__global__ __launch_bounds__(32) void cam_norm_kernel(const float* __restrict__ cs,
                                                      const float* __restrict__ ce,
                                                      const float* __restrict__ ci,
                                                      float* __restrict__ ws) {
  const int t = threadIdx.x;
  if (t >= 18) return;
  const float* src = (t < 6) ? cs : (t < 12) ? ce : ci;
  const int arr = t / 6;
  const int j = t % 6;
  float mean = 0.f;
  for (int i = 0; i < 6; ++i) mean += src[i];
  mean *= (1.0f / 6.0f);
  const float inv = 1.0f / mean;
  float mx = 0.f;
  for (int i = 0; i < 6; ++i) mx = fmaxf(mx, fabsf(src[i] * inv - 1.0f));
  const float sf = 0.3f / mx;
  const float v = src[j] * inv;
  ws[arr * 6 + j] = (sf < 1.0f) ? ((v - 1.0f) * sf + 1.0f) : v;
}

// ---------------------------------------------------------------------------
// Main fused kernel: blob generation + 9x9 depthwise conv via banded f32 WMMA
// + clip + masks running-max. One wave32 per 16x16 output tile, loop over n.
// B fragments (banded weights) are n-invariant and hoisted into 108 VGPRs.
// ---------------------------------------------------------------------------
__global__ __launch_bounds__(32) void render_kernel(
    const float* __restrict__ points,       // (NG, NPTS, 2)
    const float* __restrict__ sigmas,       // (2, NPTS)
    const float* __restrict__ exponents,    // (2, NPTS)
    const float* __restrict__ intensities,  // (2, NPTS)
    const float* __restrict__ filters,      // (NG, 9, 9)
    const float* __restrict__ cam,          // d_ws: [S x6 | E x6 | I x6]
    float* __restrict__ out)                // masks (NG,H,H) then blobs (NG,NPTS,H,H)
{
  __shared__ __align__(16) float S[STILE * SSTR];
  __shared__ __align__(16) float Wf[88];    // 9x9 filter (padded)

  const int lane = threadIdx.x;
  const int tx = blockIdx.x;                // 0..15
  const int ty = blockIdx.y;                // 0..15
  const int g  = blockIdx.z;                // 0..5
  const int b  = g / 3;

  for (int i = lane; i < 81; i += 32) Wf[i] = filters[g * 81 + i];

  const float camS = cam[g];
  const float camE = cam[6 + g];
  const float camI = cam[12 + g];

  const int m     = lane & 15;              // M (A,C,D) and N (B,C,D) index
  const int hi    = lane >> 4;              // half-wave select
  const int khalf = hi * 2;                 // K offset within 4-wide K block

  float* masks = out;
  float* blobs = out + (size_t)NG * IMG_H * IMG_H;

  __syncthreads();

  // --- hoisted: build all 54 B fragments once (n-invariant, stays in VGPRs) ---
  // B[c,x] = w[dy, c-x] for 0 <= c-x <= 8, else 0.  Per lane: rows col, col+1.
  v2f Bf[54];
#pragma unroll
  for (int kb = 0; kb < 6; ++kb) {
    const int col = kb * 4 + khalf;
    const int d0  = col - m;
    const int dc0 = min(max(d0, 0), 8);
    const int dc1 = min(max(d0 + 1, 0), 8);
    const bool v0 = (unsigned)d0 <= 8u;
    const bool v1 = (unsigned)(d0 + 1) <= 8u;
#pragma unroll
    for (int dy = 0; dy < 9; ++dy) {
      const float w0 = Wf[dy * 9 + dc0];
      const float w1 = Wf[dy * 9 + dc1];
      Bf[kb * 9 + dy].x = v0 ? w0 : 0.0f;
      Bf[kb * 9 + dy].y = v1 ? w1 : 0.0f;
    }
  }

  float runmax[8];
#pragma unroll
  for (int r = 0; r < 8; ++r) runmax[r] = 0.f;

  const int y0in = ty * TILE - HALO;
  const int x0in = tx * TILE - HALO;
  const float gstep = 2.0f / 255.0f;        // linspace(-1,1,256) step

  for (int n = 0; n < NPTS; ++n) {
    // --- per-splat parameters (uniform across the wave) ---
    float px = points[((size_t)(g * NPTS + n)) * 2 + 0];
    float py = points[((size_t)(g * NPTS + n)) * 2 + 1];
    px = fminf(fmaxf((px - 128.0f) * (1.0f / 128.0f), -1.0f), 1.0f);
    py = fminf(fmaxf((py - 128.0f) * (1.0f / 128.0f), -1.0f), 1.0f);
    const float sig   = sigmas[b * NPTS + n] * camS;
    const float e     = exponents[b * NPTS + n] * camE;
    const float inten = intensities[b * NPTS + n] * camI;
    const float inv2s2 = 1.0f / (2.0f * sig * sig);

    __syncthreads();  // WAR: previous iteration's WMMA A-loads done before rewrite

    // --- phase 1: generate 24x24 haloed blob tile (18 evals per lane) ---
#pragma unroll
    for (int i = 0; i < 18; ++i) {
      const int idx = i * 32 + lane;        // 0..575
      const int r = idx / STILE;
      const int c = idx - r * STILE;
      const int yy = y0in + r;
      const int xx = x0in + c;
      const float gx = -1.0f + gstep * (float)xx;
      const float gy = -1.0f + gstep * (float)yy;
      const float ddx = gx - px;
      const float ddy = gy - py;
      const float dst = ddx * ddx + ddy * ddy;
      const float pw  = __expf(e * __logf(dst * inv2s2));   // (dst/2s^2)^e
      const float bv  = __expf(-pw);
      const bool in = (yy >= 0) && (yy < IMG_H) && (xx >= 0) && (xx < IMG_H);
      S[r * SSTR + c] = in ? bv : 0.0f;     // SAME padding = zeros
    }
    __syncthreads();

    // --- phase 2: 9x9 depthwise conv as banded f32 GEMM on the matrix core ---
    // out[m,x] = sum_dy sum_c S[m+dy, c] * B[c,x]
    v8f acc = {0.f, 0.f, 0.f, 0.f, 0.f, 0.f, 0.f, 0.f};
#pragma unroll
    for (int kb = 0; kb < 6; ++kb) {
      const int col = kb * 4 + khalf;       // this lane's K columns: col, col+1
#pragma unroll
      for (int dy = 0; dy < 9; ++dy) {
        const v2f a = *(const v2f*)&S[(m + dy) * SSTR + col];
        acc = __builtin_amdgcn_wmma_f32_16x16x4_f32(
            false, a, false, Bf[kb * 9 + dy], (short)0, acc, false, false);
      }
    }

    // --- clip, store blobs, update masks running max ---
#pragma unroll
    for (int r = 0; r < 8; ++r) {
      const int y = ty * TILE + r + hi * 8;
      const int x = tx * TILE + m;
      const float v = fminf(fmaxf(acc[r], 0.0f), 1.0f);
      blobs[(((size_t)(g * NPTS + n)) * IMG_H + y) * IMG_H + x] = v;
      runmax[r] = fmaxf(runmax[r], v * inten);
    }
  }

#pragma unroll
  for (int r = 0; r < 8; ++r) {
    const int y = ty * TILE + r + hi * 8;
    const int x = tx * TILE + m;
    masks[((size_t)g * IMG_H + y) * IMG_H + x] = fminf(runmax[r], 1.0f);
  }
}

// ---------------------------------------------------------------------------
extern "C" void kernel_launch(void* const* d_in, const int* in_sizes, int n_in,
                              void* d_out, int out_size, void* d_ws, size_t ws_size,
                              hipStream_t stream) {
  (void)in_sizes; (void)n_in; (void)out_size; (void)ws_size;
  const float* points      = (const float*)d_in[0];
  const float* sigmas      = (const float*)d_in[1];
  const float* exponents   = (const float*)d_in[2];
  const float* intensities = (const float*)d_in[3];
  const float* cs          = (const float*)d_in[4];
  const float* ce          = (const float*)d_in[5];
  const float* ci          = (const float*)d_in[6];
  const float* filters     = (const float*)d_in[7];
  float* out = (float*)d_out;
  float* ws  = (float*)d_ws;

  cam_norm_kernel<<<dim3(1), dim3(32), 0, stream>>>(cs, ce, ci, ws);

  dim3 grid(IMG_H / TILE, IMG_H / TILE, NG);   // 16 x 16 x 6 = 1536 waves
  render_kernel<<<grid, dim3(32), 0, stream>>>(points, sigmas, exponents,
                                               intensities, filters, ws, out);
}